// ResBlock_16071767622282
// MI455X (gfx1250) — compile-verified
//
#include <hip/hip_runtime.h>
#include <cmath>
#include <cstdint>

typedef __attribute__((ext_vector_type(2))) float v2f;
typedef __attribute__((ext_vector_type(8))) float v8f;

#define HDIM 128

// ---------------------------------------------------------------------------
// Kernel 1: one wave32 per row. Compute 1/max(L1(x[row]),1e-12) and zero the
// SpMM accumulator f[row]. Lane i handles 4 consecutive floats (float4).
// ---------------------------------------------------------------------------
__global__ __launch_bounds__(256)
void k_norm_zero(const float* __restrict__ x, float* __restrict__ invn,
                 float* __restrict__ f, int n) {
  int wave = (int)((blockIdx.x * blockDim.x + threadIdx.x) >> 5);
  int lane = threadIdx.x & 31;
  if (wave >= n) return;
  float4 v = ((const float4*)(x + (size_t)wave * HDIM))[lane];
  float s = fabsf(v.x) + fabsf(v.y) + fabsf(v.z) + fabsf(v.w);
  #pragma unroll
  for (int off = 16; off > 0; off >>= 1) s += __shfl_xor(s, off, 32);
  if (lane == 0) invn[wave] = 1.0f / fmaxf(s, 1e-12f);
  float4 z = {0.f, 0.f, 0.f, 0.f};
  ((float4*)(f + (size_t)wave * HDIM))[lane] = z;
}

// ---------------------------------------------------------------------------
// Kernel 2: SpMM scatter. One wave32 per edge; lane i covers 4 columns.
// f[row] += val * (x[col] * invnorm[col])   (inf -> 0 per reference)
// x and f are L2-resident (51.2MB each << 192MB), so atomics stay on-chip.
// ---------------------------------------------------------------------------
__global__ __launch_bounds__(256)
void k_spmm(const float* __restrict__ x, const float* __restrict__ invn,
            const int* __restrict__ rows, const int* __restrict__ cols,
            const float* __restrict__ vals, float* __restrict__ f, int e) {
  int wave = (int)((blockIdx.x * blockDim.x + threadIdx.x) >> 5);
  int lane = threadIdx.x & 31;
  if (wave >= e) return;
  int r = rows[wave];
  int c = cols[wave];
  float val = vals[wave];
  float inv = invn[c];
  float4 xv = ((const float4*)(x + (size_t)c * HDIM))[lane];
  float* fr = f + (size_t)r * HDIM + lane * 4;
  float m[4] = {xv.x, xv.y, xv.z, xv.w};
  #pragma unroll
  for (int i = 0; i < 4; ++i) {
    float y = m[i] * inv;
    if (isinf(y)) y = 0.f;
    unsafeAtomicAdd(fr + i, val * y);   // global_atomic_add_f32
  }
}

// ---------------------------------------------------------------------------
// Kernel 3: fused GEMM (f @ W^T + b) with fp32 WMMA 16x16x4, then row
// L1-normalize + ReLU + residual:  out = x + relu(normalize(g)) * ts.
//
// Block = 16 rows x 128 cols of the output; 8 waves, wave w owns N-tile
// [16w, 16w+16). W (row-major, 64KB) and the 16x128 f tile are DMA'd into
// LDS with gfx1250 GLOBAL_LOAD_ASYNC_TO_LDS_B128 (no VGPR staging, tracked
// by ASYNCcnt). B[k][n] = W[n][k] -> B fragment for lane l is the float2
// W[n0+l][ka..ka+1] (ka = k for lanes 0-15, k+2 for lanes 16-31), straight
// from row-major W. A fragment is the float2 Ft[l][ka..ka+1]. 32 wmma/wave.
// ---------------------------------------------------------------------------
__global__ __launch_bounds__(256)
void k_gemm_norm(const float* __restrict__ f, const float* __restrict__ W,
                 const float* __restrict__ bias, const float* __restrict__ x,
                 const float* __restrict__ ts, float* __restrict__ out) {
  __shared__ float Ws[HDIM * HDIM];   // 64 KB: W row-major
  __shared__ float Ft[16 * HDIM];     // 8 KB: f tile, reused for GEMM output
  __shared__ float Nrm[16];

  int tid = threadIdx.x;
  size_t m0 = (size_t)blockIdx.x * 16;

  // Async DMA tile fill: 16 x 4KB sweeps for Ws, 2 x 4KB sweeps for Ft.
  // Each thread issues 18 b128 async copies (ASYNCcnt <= 63 per wave).
  {
    uint32_t wsBase = (uint32_t)(uintptr_t)&Ws[0];   // low 32 bits = LDS offset
    uint32_t ftBase = (uint32_t)(uintptr_t)&Ft[0];
    const char* wG = (const char*)W;
    const char* fG = (const char*)(f + m0 * HDIM);
    #pragma unroll
    for (int sweep = 0; sweep < 16; ++sweep) {
      uint32_t off = (uint32_t)(sweep * 4096 + tid * 16);
      uint32_t lds = wsBase + off;
      asm volatile("global_load_async_to_lds_b128 %0, %1, %2"
                   :: "v"(lds), "v"(off), "s"(wG) : "memory");
    }
    #pragma unroll
    for (int sweep = 0; sweep < 2; ++sweep) {
      uint32_t off = (uint32_t)(sweep * 4096 + tid * 16);
      uint32_t lds = ftBase + off;
      asm volatile("global_load_async_to_lds_b128 %0, %1, %2"
                   :: "v"(lds), "v"(off), "s"(fG) : "memory");
    }
    asm volatile("s_wait_asynccnt 0x0" ::: "memory");
  }
  __syncthreads();

  int w = tid >> 5;
  int lane = tid & 31;
  int l = lane & 15;      // M (A) / N (B,C,D) index within tile
  int hi = lane >> 4;     // 0: K={k,k+1}, 1: K={k+2,k+3}
  int n0 = w * 16;

  v8f acc = {};
  #pragma unroll
  for (int k = 0; k < HDIM; k += 4) {
    int ka = k + hi * 2;
    v2f a = *(const v2f*)&Ft[l * HDIM + ka];          // A: f[m0+l][ka..ka+1]
    v2f b = *(const v2f*)&Ws[(n0 + l) * HDIM + ka];   // B: W[n0+l][ka..ka+1]
    acc = __builtin_amdgcn_wmma_f32_16x16x4_f32(
        /*neg_a=*/false, a, /*neg_b=*/false, b,
        /*c_mod=*/(short)0, acc, /*reuse_a=*/false, /*reuse_b=*/false);
  }
  __syncthreads();  // all A reads done before overwriting Ft with GEMM output

  // C/D layout: VGPR r -> M = r + 8*hi, N = l. Add bias, stash in LDS.
  int col = n0 + l;
  float bb = bias[col];
  #pragma unroll
  for (int r = 0; r < 8; ++r) Ft[(r + hi * 8) * HDIM + col] = acc[r] + bb;
  __syncthreads();

  // Row L1 norms: wave w handles rows 2w (lanes 0-15) and 2w+1 (lanes 16-31);
  // each of 16 lanes sums 8 elements, then half-wave xor-reduction.
  int row = w * 2 + hi;
  float s = 0.f;
  #pragma unroll
  for (int j = 0; j < 8; ++j) s += fabsf(Ft[row * HDIM + l * 8 + j]);
  #pragma unroll
  for (int off = 8; off > 0; off >>= 1) s += __shfl_xor(s, off, 16);
  if (l == 0) Nrm[row] = s;
  __syncthreads();

  // Normalize + ReLU + residual; each thread writes 8 contiguous outputs.
  float t = ts[0];
  int base = tid * 8;
  int r2 = base >> 7;
  int c2 = base & 127;
  float inv = 1.0f / fmaxf(Nrm[r2], 1e-12f);
  size_t g = (m0 + r2) * HDIM + c2;
  #pragma unroll
  for (int j = 0; j < 8; ++j) {
    float y = Ft[r2 * HDIM + c2 + j] * inv;
    if (isinf(y)) y = 0.f;
    y = fmaxf(y, 0.f);
    out[g + j] = x[g + j] + y * t;
  }
}

// ---------------------------------------------------------------------------
extern "C" void kernel_launch(void* const* d_in, const int* in_sizes, int n_in,
                              void* d_out, int out_size, void* d_ws, size_t ws_size,
                              hipStream_t stream) {
  const float* x    = (const float*)d_in[0];
  const int*   rows = (const int*)d_in[1];
  const int*   cols = (const int*)d_in[2];
  const float* vals = (const float*)d_in[3];
  const float* W    = (const float*)d_in[4];
  const float* bias = (const float*)d_in[5];
  const float* ts   = (const float*)d_in[6];
  float* out = (float*)d_out;

  int n = in_sizes[0] / HDIM;  // 100000 nodes
  int e = in_sizes[1];         // 1600000 edges

  // Workspace: invnorm[n] | f[n*HDIM] accumulator
  float* invn = (float*)d_ws;
  size_t off = ((size_t)n * sizeof(float) + 255) & ~(size_t)255;
  float* f = (float*)((char*)d_ws + off);

  // 1) inverse L1 norms of x + zero-fill f (one wave per row)
  int blocks1 = (int)(((size_t)n * 32 + 255) / 256);
  k_norm_zero<<<blocks1, 256, 0, stream>>>(x, invn, f, n);

  // 2) edge scatter (one wave per edge)
  int blocks2 = (int)(((size_t)e * 32 + 255) / 256);
  k_spmm<<<blocks2, 256, 0, stream>>>(x, invn, rows, cols, vals, f, e);

  // 3) fused WMMA GEMM + normalize + ReLU + residual (16 rows per block)
  int blocks3 = (n + 15) / 16;
  k_gemm_norm<<<blocks3, 256, 0, stream>>>(f, W, bias, x, ts, out);
}